// Tuples_3599182594783
// MI455X (gfx1250) — compile-verified
//
#include <hip/hip_runtime.h>

// Tuples: x (B, n=512, f=128) f32 -> out (B, n*n, 2f): out[b, i*n+j] = concat(x[b,i], x[b,j])
// Store-bound: 1 GiB HBM writes (floor ~46 us @ 23.3 TB/s), 1 MiB unique reads.
// Strategy: each block stages x[b] (256 KB) into LDS once via CDNA5 async
// global->LDS DMA (ASYNCcnt path), then serves IGROUP=8 different i values
// from it -> 8x reuse of every staged byte, inner loop is nearly pure
// non-temporal b128 stores.

typedef float f4 __attribute__((ext_vector_type(4)));

#define N_OBS        512
#define F_DIM        128                   // floats per input row
#define F4_PER_ROW   32                    // float4 per input row (512 B)
#define SLAB_F4      (N_OBS * F4_PER_ROW)  // 16384 f4 = 256 KB (fits 320 KB WGP LDS)
#define BLOCK_THREADS 1024
#define IGROUP       8                     // i values served per block

__global__ __launch_bounds__(BLOCK_THREADS, 1)
void tuples_concat_kernel(const f4* __restrict__ x4,
                          f4* __restrict__ out4) {
    __shared__ f4 lds4[SLAB_F4];

    const unsigned t  = threadIdx.x;            // 0..1023
    const unsigned bg = blockIdx.x;             // b*(n/IGROUP) + ig
    const unsigned b  = bg >> 6;                // n/IGROUP == 64
    const unsigned i0 = (bg & 63u) * IGROUP;    // first i of this block's group

    // ---- Stage x[b,:,:] (256 KB) into LDS via async global->LDS (ASYNCcnt) ----
    const f4* slab = x4 + (size_t)b * SLAB_F4;
    const unsigned ldsBase = (unsigned)(size_t)&lds4[0];  // low 32 bits == LDS byte offset

    #pragma unroll
    for (int k = 0; k < SLAB_F4 / BLOCK_THREADS; ++k) {   // 16 iterations
        const unsigned idx  = (unsigned)k * BLOCK_THREADS + t;
        const unsigned loff = ldsBase + idx * 16u;
        const f4*      gp   = slab + idx;
        asm volatile("global_load_async_to_lds_b128 %0, %1, off"
                     :: "v"(loff), "v"(gp)
                     : "memory");
    }
    asm volatile("s_wait_asynccnt 0" ::: "memory");
    __syncthreads();

    // ---- Streamed stores ----
    // Thread owns fixed f4 column c of the 64-f4 output row; waves are
    // homogeneous (all-first-half or all-second-half). Out rows for the 8
    // grouped i values sit at constant 512 KB strides -> one address register
    // + immediate offsets per 8 stores.
    const unsigned c  = t & 63u;   // output f4 column
    const unsigned r0 = t >> 6;    // starting j row (0..15), stride 16

    // out row (b, i0+ii, r) == row (bg*IGROUP + ii)*512 + r
    f4* pRow0 = out4 + ((size_t)bg * IGROUP * N_OBS + r0) * 64 + c;
    const size_t iStride = (size_t)N_OBS * 64;   // 512 KB / 16 B between ii
    const size_t rStride = (size_t)16 * 64;      // 16 rows per step

    if (c < 32u) {
        // First halves: value per ii is loop-invariant over r.
        f4 v[IGROUP];
        #pragma unroll
        for (int ii = 0; ii < IGROUP; ++ii)
            v[ii] = lds4[(i0 + ii) * F4_PER_ROW + c];

        f4* p = pRow0;
        #pragma unroll 2
        for (unsigned r = r0; r < N_OBS; r += 16u) {
            #pragma unroll
            for (int ii = 0; ii < IGROUP; ++ii)
                __builtin_nontemporal_store(v[ii], p + (size_t)ii * iStride);
            p += rStride;
        }
    } else {
        // Second halves: one LDS row read serves IGROUP stores.
        const unsigned cc = c - 32u;
        f4* p = pRow0;
        #pragma unroll 2
        for (unsigned r = r0; r < N_OBS; r += 16u) {
            const f4 v = lds4[r * F4_PER_ROW + cc];
            #pragma unroll
            for (int ii = 0; ii < IGROUP; ++ii)
                __builtin_nontemporal_store(v, p + (size_t)ii * iStride);
            p += rStride;
        }
    }
}

extern "C" void kernel_launch(void* const* d_in, const int* in_sizes, int n_in,
                              void* d_out, int out_size, void* d_ws, size_t ws_size,
                              hipStream_t stream) {
    (void)n_in; (void)out_size; (void)d_ws; (void)ws_size;

    const f4* x4   = (const f4*)d_in[0];
    f4*       out4 = (f4*)d_out;

    const int B = in_sizes[0] / (N_OBS * F_DIM);          // = 4 for reference shapes

    dim3 grid((unsigned)(B * (N_OBS / IGROUP)));          // 256 blocks for B=4
    dim3 block(BLOCK_THREADS);
    tuples_concat_kernel<<<grid, block, 0, stream>>>(x4, out4);
}